// Encoder_1451698946100
// MI455X (gfx1250) — compile-verified
//
#include <hip/hip_runtime.h>
#include <stdint.h>

// ---------------------------------------------------------------------------
// Encoder: out = relu(x + w * segment_sum(ew[:,None] * x[src], dst))
//   x:  [N,64] f32, edge_index: [2,E] int32, ew: [E] f32, w: [1] f32
// Memory/atomic-bound scatter: no matmul structure -> WMMA inapplicable.
// CDNA5 paths: global_load_async_to_lds_b128 (ASYNCcnt) double-buffered edge
// staging with NT hint, s_wait_asynccnt, global_prefetch_b8 row prefetch,
// native fp32 agent-scope atomics, wave32 16-lane row groups, and 192MB-L2
// residency of both x (25.6MB) and out (25.6MB).
// ---------------------------------------------------------------------------

#define FDIM    64
#define CHUNK   64     // edges staged per block-iteration
#define PASSES  4      // CHUNK / 16 edges processed per pass
#define BLK     256    // threads per block (16 edges x 16 lanes per pass)

// Never launched: placed first in the module so the disasm snippet shows the
// exact lowering of the fp32 atomic (want: global_atomic_add_f32, not a CAS
// loop) and of __builtin_prefetch (want: global_prefetch_b8).
__global__ void k_probe(float* p, float v) {
  __builtin_prefetch(p + 1024, 0, 3);
  __hip_atomic_fetch_add(p, v, __ATOMIC_RELAXED, __HIP_MEMORY_SCOPE_AGENT);
}

__device__ __forceinline__ void async_ld_b128(unsigned lds_addr, unsigned byte_off,
                                              unsigned long long gbase) {
  // GVS mode: saddr = 64-bit SGPR base, vaddr = 32-bit byte offset,
  // vdst = LDS byte address. Edge stream is read-once -> non-temporal.
  asm volatile("global_load_async_to_lds_b128 %0, %1, %2 th:TH_LOAD_NT"
               :: "v"(lds_addr), "v"(byte_off), "s"(gbase)
               : "memory");
}

__device__ __forceinline__ void async_wait0() {
  asm volatile("s_wait_asynccnt 0x0" ::: "memory");
}

__device__ __forceinline__ unsigned lds_off(const void* p) {
  // Generic pointers to LDS carry the group-segment offset in the low 32 bits.
  return (unsigned)(size_t)p;
}

__device__ __forceinline__ void atomic_add_f32(float* p, float v) {
  __hip_atomic_fetch_add(p, v, __ATOMIC_RELAXED, __HIP_MEMORY_SCOPE_AGENT);
}

// ---------------------------------------------------------------------------
__global__ __launch_bounds__(BLK) void k_scatter(
    const float* __restrict__ x,
    const int*   __restrict__ esrc,
    const int*   __restrict__ edst,
    const float* __restrict__ ew,
    float*       __restrict__ out,
    int nEdges)
{
  __shared__ int   s_src[2][CHUNK];
  __shared__ int   s_dst[2][CHUNK];
  __shared__ float s_ew [2][CHUNK];

  const int tid     = threadIdx.x;
  const int nFull   = nEdges / CHUNK;                 // fully-staged chunks
  const int nChunks = (nEdges + CHUNK - 1) / CHUNK;   // incl. partial tail

  // Guard-free staging of one FULL 64-edge chunk into LDS buffer b.
  // Lanes 0..15 -> src (b128 = 4 ints each), 16..31 -> dst, 32..47 -> ew.
  auto stage = [&](int c, int b) {
    const int j = tid & 15;
    const unsigned off = (unsigned)(c * CHUNK + j * 4) * 4u;
    if (tid < 16) {
      async_ld_b128(lds_off(&s_src[b][j * 4]), off, (unsigned long long)(size_t)esrc);
    } else if (tid < 32) {
      async_ld_b128(lds_off(&s_dst[b][j * 4]), off, (unsigned long long)(size_t)edst);
    } else if (tid < 48) {
      async_ld_b128(lds_off(&s_ew[b][j * 4]), off, (unsigned long long)(size_t)ew);
    }
  };

  int c = blockIdx.x;
  if (c < nFull) stage(c, 0);

  const int jj   = tid >> 4;   // edge slot within a 16-edge pass (0..15)
  const int lane = tid & 15;   // float4 quad within the 64-wide row

  for (int it = 0; c < nChunks; c += gridDim.x, ++it) {
    if (c < nFull) {
      const int b = it & 1;

      async_wait0();       // staged metadata for chunk c has landed in LDS
      __syncthreads();     // visible to all 8 waves of the workgroup

      const int cn = c + gridDim.x;
      if (cn < nFull) stage(cn, b ^ 1);   // DMA next chunk into other buffer

      // Prefetch all 64 gather rows of this chunk (4 lanes x 64B per 256B
      // row) so passes overlap atomics with row fetch.
      {
        const int slot = tid >> 2;        // 0..63
        const int seg  = tid & 3;         // 64B segment within the row
        const char* pf =
            (const char*)(x + (size_t)s_src[b][slot] * FDIM) + seg * 64;
        __builtin_prefetch(pf, 0, 3);     // -> global_prefetch_b8
      }

      #pragma unroll
      for (int p = 0; p < PASSES; ++p) {
        const int slot = p * 16 + jj;
        const int   srcn = s_src[b][slot];   // LDS broadcast reads
        const int   dstn = s_dst[b][slot];
        const float wgt  = s_ew [b][slot];

        // 16 lanes x 16B = one fully-coalesced 256B row gather (L2-resident).
        const float4 v = ((const float4*)(x + (size_t)srcn * FDIM))[lane];
        float* o = out + (size_t)dstn * FDIM + lane * 4;

        atomic_add_f32(o + 0, wgt * v.x);
        atomic_add_f32(o + 1, wgt * v.y);
        atomic_add_f32(o + 2, wgt * v.z);
        atomic_add_f32(o + 3, wgt * v.w);
      }
      // No trailing barrier: next iteration's wait+barrier orders buffer
      // reuse (buffer b is overwritten by the next iteration's stage, after its barrier).
    } else {
      // At most one partial tail chunk: cold direct-from-global path.
      for (int p = 0; p < PASSES; ++p) {
        const int e = c * CHUNK + p * 16 + jj;
        if (e < nEdges) {
          const int   srcn = esrc[e];
          const int   dstn = edst[e];
          const float wgt  = ew[e];
          const float4 v = ((const float4*)(x + (size_t)srcn * FDIM))[lane];
          float* o = out + (size_t)dstn * FDIM + lane * 4;
          atomic_add_f32(o + 0, wgt * v.x);
          atomic_add_f32(o + 1, wgt * v.y);
          atomic_add_f32(o + 2, wgt * v.z);
          atomic_add_f32(o + 3, wgt * v.w);
        }
      }
    }
  }
}

// ---------------------------------------------------------------------------
__global__ void k_zero(float4* __restrict__ out, int n4) {
  int i = blockIdx.x * blockDim.x + threadIdx.x;
  if (i < n4) out[i] = make_float4(0.f, 0.f, 0.f, 0.f);
}

// ---------------------------------------------------------------------------
__global__ void k_combine(const float4* __restrict__ x,
                          const float*  __restrict__ wptr,
                          float4*       __restrict__ out, int n4) {
  int i = blockIdx.x * blockDim.x + threadIdx.x;
  if (i < n4) {
    const float  w = wptr[0];
    const float4 a = x[i];
    const float4 g = out[i];
    float4 r;
    r.x = fmaxf(fmaf(w, g.x, a.x), 0.f);
    r.y = fmaxf(fmaf(w, g.y, a.y), 0.f);
    r.z = fmaxf(fmaf(w, g.z, a.z), 0.f);
    r.w = fmaxf(fmaf(w, g.w, a.w), 0.f);
    out[i] = r;
  }
}

// ---------------------------------------------------------------------------
extern "C" void kernel_launch(void* const* d_in, const int* in_sizes, int n_in,
                              void* d_out, int out_size, void* d_ws, size_t ws_size,
                              hipStream_t stream) {
  const float* x    = (const float*)d_in[0];
  const int*   eidx = (const int*)  d_in[1];   // [2, E] int32
  const float* ew   = (const float*)d_in[2];
  const float* w    = (const float*)d_in[3];
  float*       out  = (float*)d_out;

  const int E = in_sizes[2];          // edge count from edge_weights
  const int N = in_sizes[0] / FDIM;   // node count

  const int* esrc = eidx;             // row 0: source nodes
  const int* edst = eidx + E;         // row 1: aggregation targets

  const int n4 = N * (FDIM / 4);

  // 1) zero the accumulator (d_out is poisoned by the harness)
  k_zero<<<(n4 + 255) / 256, 256, 0, stream>>>((float4*)out, n4);

  // 2) edge scatter-add with async-LDS staged edge stream + row prefetch
  const int nChunks = (E + CHUNK - 1) / CHUNK;
  const int grid    = nChunks < 2048 ? nChunks : 2048;
  k_scatter<<<grid, BLK, 0, stream>>>(x, esrc, edst, ew, out, E);

  // 3) out = relu(x + w * agg)
  k_combine<<<(n4 + 255) / 256, 256, 0, stream>>>((const float4*)x, w,
                                                  (float4*)out, n4);
}